// Pointnet2Backbone_tflite_33251636806263
// MI455X (gfx1250) — compile-verified
//
#include <hip/hip_runtime.h>
#include <hip/hip_bf16.h>

// Problem constants (from reference)
#define BATCH   8
#define NPTS    20000
#define NCH     5
#define NPOINT  1024
#define NSAMPLE 64
#define RADIUS  0.2f

typedef float v2f __attribute__((ext_vector_type(2)));
typedef float v8f __attribute__((ext_vector_type(8)));

// ---------------------------------------------------------------------------
// Kernel 1: farthest point sampling. One block per batch, 256 threads (8
// waves, wave32). mindist lives in global scratch (L2-resident, 80KB/batch).
// Writes indices as float into d_out's inds region.
// ---------------------------------------------------------------------------
__global__ __launch_bounds__(256) void fps_kernel(const float* __restrict__ pc,
                                                  float* __restrict__ inds_f,
                                                  float* __restrict__ mind_ws) {
    const int b   = blockIdx.x;
    const int tid = threadIdx.x;
    float* mind = mind_ws + (size_t)b * NPTS;
    const float* base = pc + (size_t)b * NPTS * NCH;

    __shared__ float redv[8];
    __shared__ int   redi[8];
    __shared__ float lastx, lasty, lastz;
    __shared__ int   lastIdx;

    if (tid == 0) {
        lastIdx = 0;
        inds_f[b * NPOINT] = 0.0f;
    }
    __syncthreads();

    for (int t = 1; t < NPOINT; ++t) {
        if (tid == 0) {
            const float* p = base + (size_t)lastIdx * NCH;
            lastx = p[0]; lasty = p[1]; lastz = p[2];
        }
        __syncthreads();
        float lx = lastx, ly = lasty, lz = lastz;

        float bestd = -1.0f;
        int   besti = 0;
        for (int i = tid; i < NPTS; i += 256) {
            const float* p = base + (size_t)i * NCH;
            float dx = p[0] - lx, dy = p[1] - ly, dz = p[2] - lz;
            float d = dx * dx + dy * dy + dz * dz;
            float m = (t == 1) ? d : fminf(mind[i], d);  // t==1 fully rewrites scratch
            mind[i] = m;
            if (m > bestd) { bestd = m; besti = i; }
        }
        // wave32 reduction (argmax, prefer lower index on tie)
        #pragma unroll
        for (int off = 16; off > 0; off >>= 1) {
            float od = __shfl_down(bestd, off, 32);
            int   oi = __shfl_down(besti, off, 32);
            if (od > bestd || (od == bestd && oi < besti)) { bestd = od; besti = oi; }
        }
        int w = tid >> 5;
        if ((tid & 31) == 0) { redv[w] = bestd; redi[w] = besti; }
        __syncthreads();
        if (tid == 0) {
            #pragma unroll
            for (int j = 1; j < 8; ++j) {
                if (redv[j] > bestd || (redv[j] == bestd && redi[j] < besti)) {
                    bestd = redv[j]; besti = redi[j];
                }
            }
            lastIdx = besti;
            inds_f[b * NPOINT + t] = (float)besti;
        }
        __syncthreads();
    }
}

// ---------------------------------------------------------------------------
// Kernel 2: gather new_xyz = xyz[inds]
// ---------------------------------------------------------------------------
__global__ __launch_bounds__(256) void gather_newxyz(const float* __restrict__ pc,
                                                     const float* __restrict__ inds_f,
                                                     float* __restrict__ newxyz) {
    int g = blockIdx.x * blockDim.x + threadIdx.x;
    if (g >= BATCH * NPOINT) return;
    int b   = g >> 10;
    int idx = (int)inds_f[g];
    const float* p = pc + ((size_t)b * NPTS + idx) * NCH;
    newxyz[g * 3 + 0] = p[0];
    newxyz[g * 3 + 1] = p[1];
    newxyz[g * 3 + 2] = p[2];
}

// ---------------------------------------------------------------------------
// Uniform 8-way select: lane L (0..15) picks ballot word (L&7), then its half.
// Pure v_cndmask tree, no LDS, no EXEC manipulation.
// ---------------------------------------------------------------------------
__device__ __forceinline__ unsigned row_mask16(unsigned m0, unsigned m1, unsigned m2,
                                               unsigned m3, unsigned m4, unsigned m5,
                                               unsigned m6, unsigned m7, int lane) {
    unsigned t01 = (lane & 1) ? m1 : m0;
    unsigned t23 = (lane & 1) ? m3 : m2;
    unsigned t45 = (lane & 1) ? m5 : m4;
    unsigned t67 = (lane & 1) ? m7 : m6;
    unsigned a   = (lane & 2) ? t23 : t01;
    unsigned bb  = (lane & 2) ? t67 : t45;
    unsigned t   = (lane & 4) ? bb : a;
    return ((lane & 8) ? (t >> 16) : t) & 0xFFFFu;
}

// ---------------------------------------------------------------------------
// Kernel 3: fused ball query + grouping. One wave per 16-query tile.
// Distance tiles via V_WMMA_F32_16X16X4_F32 (x2 per iteration, 32 pts):
//   A row m = (-2qx, -2qy, -2qz, 1)
//   B col n = ( px,  py,  pz, |p|^2)
//   D[m][n] = |p|^2 - 2 q.p ; hit iff D <= r^2 - |q|^2 (per-row threshold,
//   hoisted into registers). Row hit masks extracted from wave ballots with a
//   register select tree; lanes 0-15 append first-NSAMPLE indices in ascending
//   order (CUDA ball-query semantics). No LDS/barriers in the hot loop.
// ---------------------------------------------------------------------------
__global__ __launch_bounds__(32) void ball_group(const float* __restrict__ pc,
                                                 const float* __restrict__ newxyz,
                                                 float* __restrict__ grouped) {
    const float r2 = RADIUS * RADIUS;
    const int lane  = threadIdx.x;
    const bool hi   = lane >= 16;
    const int b     = blockIdx.y;
    const int mtile = blockIdx.x * 16;
    const int row   = lane & 15;
    const int m     = mtile + row;

    const float* base = pc + (size_t)b * NPTS * NCH;
    const float* q    = newxyz + ((size_t)b * NPOINT + m) * 3;
    float qx = q[0], qy = q[1], qz = q[2];

    __shared__ float th[16];
    __shared__ float qpt[16][3];
    __shared__ int   ballIdx[16 * NSAMPLE];

    if (lane < 16) {
        th[lane]     = r2 - (qx * qx + qy * qy + qz * qz);
        qpt[lane][0] = qx; qpt[lane][1] = qy; qpt[lane][2] = qz;
    }
    __syncthreads();

    // Hoist per-row thresholds into registers (loop-invariant).
    const float* thp = th + (hi ? 8 : 0);
    float tr0 = thp[0], tr1 = thp[1], tr2 = thp[2], tr3 = thp[3];
    float tr4 = thp[4], tr5 = thp[5], tr6 = thp[6], tr7 = thp[7];

    // A-matrix operand (16x4 f32): VGPR0 = K0 (lanes 0-15) | K2 (16-31),
    //                              VGPR1 = K1 (lanes 0-15) | K3 (16-31)
    v2f a;
    a.x = hi ? (-2.0f * qz) : (-2.0f * qx);
    a.y = hi ? 1.0f         : (-2.0f * qy);

    int cnt = 0, firstIdx = 0;

    for (int n0 = 0; n0 < NPTS; n0 += 32) {   // 20000 % 32 == 0
        const float* p0 = base + (size_t)(n0 + row) * NCH;
        const float* p1 = base + (size_t)(n0 + 16 + row) * NCH;
        float px0 = p0[0], py0 = p0[1], pz0 = p0[2];
        float px1 = p1[0], py1 = p1[1], pz1 = p1[2];

        v2f b0, b1;
        b0.x = hi ? pz0 : px0;
        b0.y = hi ? (px0 * px0 + py0 * py0 + pz0 * pz0) : py0;
        b1.x = hi ? pz1 : px1;
        b1.y = hi ? (px1 * px1 + py1 * py1 + pz1 * pz1) : py1;

        v8f z = {};
        v8f d0 = __builtin_amdgcn_wmma_f32_16x16x4_f32(
            false, a, false, b0, (short)0, z, false, false);
        v8f d1 = __builtin_amdgcn_wmma_f32_16x16x4_f32(
            false, a, false, b1, (short)0, z, false, false);

        // Wave ballots: ballot(d[r] <= tr_r) packs rows r (bits 0-15) and r+8
        // (bits 16-31).
        unsigned s00 = __builtin_amdgcn_ballot_w32(d0[0] <= tr0);
        unsigned s01 = __builtin_amdgcn_ballot_w32(d0[1] <= tr1);
        unsigned s02 = __builtin_amdgcn_ballot_w32(d0[2] <= tr2);
        unsigned s03 = __builtin_amdgcn_ballot_w32(d0[3] <= tr3);
        unsigned s04 = __builtin_amdgcn_ballot_w32(d0[4] <= tr4);
        unsigned s05 = __builtin_amdgcn_ballot_w32(d0[5] <= tr5);
        unsigned s06 = __builtin_amdgcn_ballot_w32(d0[6] <= tr6);
        unsigned s07 = __builtin_amdgcn_ballot_w32(d0[7] <= tr7);
        unsigned s10 = __builtin_amdgcn_ballot_w32(d1[0] <= tr0);
        unsigned s11 = __builtin_amdgcn_ballot_w32(d1[1] <= tr1);
        unsigned s12 = __builtin_amdgcn_ballot_w32(d1[2] <= tr2);
        unsigned s13 = __builtin_amdgcn_ballot_w32(d1[3] <= tr3);
        unsigned s14 = __builtin_amdgcn_ballot_w32(d1[4] <= tr4);
        unsigned s15 = __builtin_amdgcn_ballot_w32(d1[5] <= tr5);
        unsigned s16 = __builtin_amdgcn_ballot_w32(d1[6] <= tr6);
        unsigned s17 = __builtin_amdgcn_ballot_w32(d1[7] <= tr7);

        if (lane < 16 && cnt < NSAMPLE) {
            unsigned lo = row_mask16(s00, s01, s02, s03, s04, s05, s06, s07, lane);
            unsigned up = row_mask16(s10, s11, s12, s13, s14, s15, s16, s17, lane);
            unsigned msk = lo | (up << 16);   // 32 candidate points, ascending
            while (msk && cnt < NSAMPLE) {
                int j   = __builtin_ctz(msk);
                int idx = n0 + j;
                if (cnt == 0) firstIdx = idx;
                ballIdx[lane * NSAMPLE + cnt] = idx;
                ++cnt;
                msk &= msk - 1;
            }
        }
        unsigned done = __builtin_amdgcn_ballot_w32(lane >= 16 || cnt >= NSAMPLE);
        if (done == 0xFFFFFFFFu) break;
    }

    if (lane < 16) {
        for (int k = cnt; k < NSAMPLE; ++k) ballIdx[lane * NSAMPLE + k] = firstIdx;
    }
    __syncthreads();

    // Grouping epilogue: 16*64 gathered points, 4 output channels each.
    for (int e = lane; e < 16 * NSAMPLE; e += 32) {
        int mq  = e >> 6;
        int k   = e & (NSAMPLE - 1);
        int idx = ballIdx[e];
        const float* p = base + (size_t)idx * NCH;
        float gx = (p[0] - qpt[mq][0]) * (1.0f / RADIUS);
        float gy = (p[1] - qpt[mq][1]) * (1.0f / RADIUS);
        float gz = (p[2] - qpt[mq][2]) * (1.0f / RADIUS);
        float ft = p[4];
        float4 o = make_float4(gx, gy, gz, ft);
        *(float4*)(grouped + (((size_t)b * NPOINT + mtile + mq) * NSAMPLE + k) * 4) = o;
    }
}

// ---------------------------------------------------------------------------
// d_out layout (floats): [0,24576) new_xyz | [24576,32768) inds (as float) |
//                        [32768,2129920) grouped_features
// d_ws: 8*20000 floats of FPS mindist scratch (fully rewritten each call).
// ---------------------------------------------------------------------------
extern "C" void kernel_launch(void* const* d_in, const int* in_sizes, int n_in,
                              void* d_out, int out_size, void* d_ws, size_t ws_size,
                              hipStream_t stream) {
    const float* pc = (const float*)d_in[0];
    float* out     = (float*)d_out;
    float* newxyz  = out;                       // 8*1024*3
    float* inds_f  = out + BATCH * NPOINT * 3;  // 8*1024
    float* grouped = inds_f + BATCH * NPOINT;   // 8*1024*64*4
    float* mind    = (float*)d_ws;              // 8*20000

    fps_kernel<<<BATCH, 256, 0, stream>>>(pc, inds_f, mind);
    gather_newxyz<<<(BATCH * NPOINT + 255) / 256, 256, 0, stream>>>(pc, inds_f, newxyz);
    dim3 g(NPOINT / 16, BATCH);
    ball_group<<<g, 32, 0, stream>>>(pc, newxyz, grouped);
}